// GCN_9311489098471
// MI455X (gfx1250) — compile-verified
//
#include <hip/hip_runtime.h>
#include <hip/hip_bf16.h>

// ---------------------------------------------------------------------------
// GIN (3 conv layers + BN) + global_add_pool + MLP head, for MI455X (gfx1250).
// Memory/atomic bound workload (~1.4 GB traffic, ~1.3 GFLOP dense), so the
// GEMMs use full-precision V_WMMA_F32_16X16X4_F32 (wave32 WMMA, fp32 in/out)
// to match the fp32 reference numerics -- the matrix pipe is nowhere near the
// bottleneck, HBM (23.3 TB/s) is.
// ---------------------------------------------------------------------------

typedef __attribute__((ext_vector_type(2))) float v2f;
typedef __attribute__((ext_vector_type(8))) float v8f;

#define N_NODES 50000
#define N_EDGES 800000
#define N_GRAPHS 256
#define HID 64
#define BN_EPS 1e-5f

// ---------------------------------------------------------------- zero fill
__global__ __launch_bounds__(256) void zero_f32(float* __restrict__ p, int n) {
    int i = blockIdx.x * 256 + threadIdx.x;
    if (i < n) p[i] = 0.0f;
}

// ------------------------------------------------------- edge scatter (agg)
// agg[dst] += w_e * h[src]   ; one thread per (edge, feature)
__global__ __launch_bounds__(256) void gin_scatter(const float* __restrict__ h,
                                                   const int* __restrict__ ei,
                                                   const float* __restrict__ ew,
                                                   float* __restrict__ agg,
                                                   int nedges) {
    int t = blockIdx.x * 256 + threadIdx.x;
    int e = t >> 6;
    int f = t & 63;
    if (e < nedges) {
        int src = ei[e];
        int dst = ei[nedges + e];
        float w = ew[e];
        atomicAdd(&agg[(size_t)dst * HID + f], w * h[(size_t)src * HID + f]);
    }
}

// ----------------------------------------------------- fused GEMM + epilogue
// pre = (h + agg) @ W + bias  [; relu]  and accumulate per-column sum/sumsq.
// 4 waves per block; each wave computes one 16x64 output tile with
// 16 (K-chunks) x 4 (N-tiles) = 64 v_wmma_f32_16x16x4_f32 instructions.
//
// A layout (32-bit 16x4, ISA 7.12.2): lanes 0-15 hold M=lane, VGPR0=K0,
// VGPR1=K1; lanes 16-31 hold M=lane-16, VGPR0=K2, VGPR1=K3.
// B layout mirrors with N instead of M. C/D: VGPR v holds row v+8*(lane>=16),
// col = lane&15 (+16*tile).
__global__ __launch_bounds__(128) void gin_gemm_wmma(const float* __restrict__ hin,
                                                     const float* __restrict__ agg,
                                                     const float* __restrict__ W,
                                                     const float* __restrict__ bias,
                                                     float* __restrict__ pre,
                                                     float* __restrict__ stats,
                                                     int nrows, int do_relu) {
    __shared__ float sW[HID * HID];
    __shared__ float sSum[HID];
    __shared__ float sSq[HID];

    const int tid  = threadIdx.x;   // 0..127
    const int wave = tid >> 5;      // 0..3
    const int lane = tid & 31;

    for (int i = tid; i < HID * HID; i += 128) sW[i] = W[i];
    if (tid < HID) { sSum[tid] = 0.0f; sSq[tid] = 0.0f; }
    __syncthreads();

    const int tile = blockIdx.x * 4 + wave;  // 16-row tile
    const int row0 = tile * 16;
    if (row0 < nrows) {                      // wave-uniform branch: EXEC all 1s
        const int mrow  = lane & 15;
        const int khalf = (lane >> 4) * 2;   // 0 or 2
        const int row   = row0 + mrow;
        const float* hr = hin + (size_t)row * HID;
        const float* ar = agg + (size_t)row * HID;

        v2f a[16];
#pragma unroll
        for (int j = 0; j < 16; ++j) {
            int k = 4 * j + khalf;
            v2f av = { hr[k] + ar[k], hr[k + 1] + ar[k + 1] };
            a[j] = av;
        }

        v8f acc0 = {}, acc1 = {}, acc2 = {}, acc3 = {};
        const int ncol = lane & 15;
#pragma unroll
        for (int j = 0; j < 16; ++j) {
            int k = 4 * j + khalf;
            const float* w0 = sW + k * HID;
            const float* w1 = sW + (k + 1) * HID;
            v2f b0 = { w0[ncol],      w1[ncol]      };
            v2f b1 = { w0[ncol + 16], w1[ncol + 16] };
            v2f b2 = { w0[ncol + 32], w1[ncol + 32] };
            v2f b3 = { w0[ncol + 48], w1[ncol + 48] };
            acc0 = __builtin_amdgcn_wmma_f32_16x16x4_f32(false, a[j], false, b0,
                                                         (short)0, acc0, false, false);
            acc1 = __builtin_amdgcn_wmma_f32_16x16x4_f32(false, a[j], false, b1,
                                                         (short)0, acc1, false, false);
            acc2 = __builtin_amdgcn_wmma_f32_16x16x4_f32(false, a[j], false, b2,
                                                         (short)0, acc2, false, false);
            acc3 = __builtin_amdgcn_wmma_f32_16x16x4_f32(false, a[j], false, b3,
                                                         (short)0, acc3, false, false);
        }

        const int hi = (lane >> 4) * 8;
#pragma unroll
        for (int v = 0; v < 8; ++v) {
            int r = row0 + hi + v;
            float vals[4] = { acc0[v], acc1[v], acc2[v], acc3[v] };
#pragma unroll
            for (int t = 0; t < 4; ++t) {
                int c = ncol + 16 * t;
                float x = vals[t] + bias[c];
                if (do_relu) x = fmaxf(x, 0.0f);
                pre[(size_t)r * HID + c] = x;
                atomicAdd(&sSum[c], x);       // ds_add_f32
                atomicAdd(&sSq[c], x * x);
            }
        }
    }
    __syncthreads();
    if (tid < HID) {
        atomicAdd(&stats[tid], sSum[tid]);
        atomicAdd(&stats[HID + tid], sSq[tid]);
    }
}

// --------------------------------------------------------------- BN finalize
__global__ void bn_finalize(const float* __restrict__ stats,
                            const float* __restrict__ gamma,
                            const float* __restrict__ beta,
                            float* __restrict__ ss, float inv_n) {
    int c = threadIdx.x;  // 64 threads
    float mu  = stats[c] * inv_n;
    float var = stats[HID + c] * inv_n - mu * mu;
    float inv = rsqrtf(var + BN_EPS);
    float sc  = gamma[c] * inv;
    ss[c] = sc;
    ss[HID + c] = beta[c] - mu * sc;
}

// ------------------------------------------------------------------ BN apply
__global__ __launch_bounds__(256) void bn_apply(const float* __restrict__ pre,
                                                const float* __restrict__ ss,
                                                float* __restrict__ out, int n) {
    int i = blockIdx.x * 256 + threadIdx.x;
    if (i < n) {
        int c = i & 63;
        out[i] = pre[i] * ss[c] + ss[HID + c];
    }
}

// ------------------------------------------------------------ global add pool
__global__ __launch_bounds__(256) void pool_sum(const float* __restrict__ h,
                                                const int* __restrict__ batch,
                                                float* __restrict__ pooled,
                                                int nnodes) {
    int t = blockIdx.x * 256 + threadIdx.x;
    int node = t >> 6;
    int f = t & 63;
    if (node < nnodes) {
        int g = batch[node];
        atomicAdd(&pooled[(size_t)g * HID + f], h[(size_t)node * HID + f]);
    }
}

// ------------------------------------------------------------------ MLP head
// one 256-thread block: graph per thread; fcW1/fcW2/fcb1 in LDS (broadcasts).
__global__ __launch_bounds__(256) void mlp_head(const float* __restrict__ pooled,
                                                const float* __restrict__ fcW1,
                                                const float* __restrict__ fcb1,
                                                const float* __restrict__ fcW2,
                                                const float* __restrict__ fcb2,
                                                float* __restrict__ out) {
    __shared__ float sW1[HID * HID];
    __shared__ float sb1[HID];
    __shared__ float sW2[HID];
    int tid = threadIdx.x;
    for (int i = tid; i < HID * HID; i += 256) sW1[i] = fcW1[i];
    if (tid < HID) { sb1[tid] = fcb1[tid]; sW2[tid] = fcW2[tid]; }
    __syncthreads();

    float p[HID];
#pragma unroll
    for (int k = 0; k < HID; ++k) p[k] = fmaxf(pooled[(size_t)tid * HID + k], 0.0f);

    float o = 0.0f;
    for (int n = 0; n < HID; ++n) {
        float acc = sb1[n];
#pragma unroll
        for (int k = 0; k < HID; ++k) acc += p[k] * sW1[k * HID + n];
        o += fmaxf(acc, 0.0f) * sW2[n];
    }
    out[tid] = o + fcb2[0];
}

// ---------------------------------------------------------------------------
extern "C" void kernel_launch(void* const* d_in, const int* in_sizes, int n_in,
                              void* d_out, int out_size, void* d_ws, size_t ws_size,
                              hipStream_t stream) {
    const float* x     = (const float*)d_in[0];
    const int*   ei    = (const int*)d_in[1];   // int32 (JAX x64 disabled)
    const float* ew    = (const float*)d_in[2];
    const int*   batch = (const int*)d_in[3];
    const float* Wl[3]  = { (const float*)d_in[4], (const float*)d_in[6], (const float*)d_in[8] };
    const float* bl[3]  = { (const float*)d_in[5], (const float*)d_in[7], (const float*)d_in[9] };
    const float* fcW1  = (const float*)d_in[10];
    const float* fcb1  = (const float*)d_in[11];
    const float* fcW2  = (const float*)d_in[12];
    const float* fcb2  = (const float*)d_in[13];
    const float* gm[3] = { (const float*)d_in[14], (const float*)d_in[16], (const float*)d_in[18] };
    const float* bt[3] = { (const float*)d_in[15], (const float*)d_in[17], (const float*)d_in[19] };
    float* out = (float*)d_out;

    const size_t NF = (size_t)N_NODES * HID;   // 3.2M floats
    float* bufH   = (float*)d_ws;              // normalized features (ping)
    float* bufAgg = bufH + NF;                 // edge aggregation
    float* bufPre = bufAgg + NF;               // pre-BN activations
    float* stats  = bufPre + NF;               // [64] sum, [64] sumsq
    float* ss     = stats + 2 * HID;           // [64] scale, [64] shift
    float* pooled = ss + 2 * HID;              // [256 x 64]

    const int elemBlocks    = (int)((NF + 255) / 256);                       // 12500
    const int scatterBlocks = (N_EDGES * HID + 255) / 256;                   // 200000
    const int gemmBlocks    = ((N_NODES + 15) / 16 + 3) / 4;                 // 782
    const int poolBlocks    = (N_NODES * HID + 255) / 256;                   // 12500

    for (int l = 0; l < 3; ++l) {
        const float* hin = (l == 0) ? x : bufH;
        zero_f32<<<elemBlocks, 256, 0, stream>>>(bufAgg, (int)NF);
        zero_f32<<<1, 256, 0, stream>>>(stats, 2 * HID);
        gin_scatter<<<scatterBlocks, 256, 0, stream>>>(hin, ei, ew, bufAgg, N_EDGES);
        gin_gemm_wmma<<<gemmBlocks, 128, 0, stream>>>(hin, bufAgg, Wl[l], bl[l],
                                                      bufPre, stats, N_NODES,
                                                      (l < 2) ? 1 : 0);
        bn_finalize<<<1, HID, 0, stream>>>(stats, gm[l], bt[l], ss, 1.0f / (float)N_NODES);
        bn_apply<<<elemBlocks, 256, 0, stream>>>(bufPre, ss, bufH, (int)NF);
    }

    zero_f32<<<(N_GRAPHS * HID + 255) / 256, 256, 0, stream>>>(pooled, N_GRAPHS * HID);
    pool_sum<<<poolBlocks, 256, 0, stream>>>(bufH, batch, pooled, N_NODES);
    mlp_head<<<1, 256, 0, stream>>>(pooled, fcW1, fcb1, fcW2, fcb2, out);
}